// MessageFunctionForEvent_592705487031
// MI455X (gfx1250) — compile-verified
//
#include <hip/hip_runtime.h>
#include <hip/hip_bf16.h>

// Problem constants (match reference)
#define BB   16
#define FF   128      // F == M
#define MD   128
#define NN   20000
#define NT   80       // N-columns per block (20000 / 80 = 250 tiles exactly)
#define SK   132      // padded K stride (dwords) in LDS, conflict-free b64 loads

typedef __attribute__((ext_vector_type(2))) float v2f;
typedef __attribute__((ext_vector_type(4))) float v4f;
typedef __attribute__((ext_vector_type(8))) float v8f;

// ---------------- prep kernels (tiny) ----------------

// A1[k][f] = sum_m W_resize[k][m]       * W_e2m[m][f]
// A2[k][f] = sum_m W_resize[k][128 + m] * W_n2m[m][f]
__global__ __launch_bounds__(256) void prep_A(const float* __restrict__ Wr,
                                              const float* __restrict__ We2m,
                                              const float* __restrict__ Wn2m,
                                              float* __restrict__ A1,
                                              float* __restrict__ A2) {
    int idx = blockIdx.x * 256 + threadIdx.x;   // 0 .. 32767
    int g   = idx >> 14;                        // 0 -> A1, 1 -> A2
    int kf  = idx & 16383;
    int k   = kf >> 7;
    int f   = kf & 127;
    const float* wsel = Wr + k * (3 * MD) + g * MD;
    const float* wx   = g ? Wn2m : We2m;
    float s = 0.f;
    #pragma unroll 8
    for (int m = 0; m < MD; ++m) s += wsel[m] * wx[m * FF + f];
    (g ? A2 : A1)[k * FF + f] = s;
}

// nv[b][m] = h_v[b] . W_n2m[m] + b_n2m[m]
__global__ __launch_bounds__(256) void prep_nv(const float* __restrict__ h_v,
                                               const float* __restrict__ Wn2m,
                                               const float* __restrict__ b_n2m,
                                               float* __restrict__ nv) {
    int idx = blockIdx.x * 256 + threadIdx.x;   // 0 .. 2047
    int b = idx >> 7, m = idx & 127;
    float s = b_n2m[m];
    #pragma unroll 8
    for (int f = 0; f < FF; ++f) s += h_v[b * FF + f] * Wn2m[m * FF + f];
    nv[idx] = s;
}

// cvec[b][k] = b_resize[k] + sum_m ( Wa[k][m]*b_e2m[m] + Wb[k][m]*b_n2m[m] + Wc[k][m]*nv[b][m] )
__global__ __launch_bounds__(256) void prep_cvec(const float* __restrict__ Wr,
                                                 const float* __restrict__ b_e2m,
                                                 const float* __restrict__ b_n2m,
                                                 const float* __restrict__ b_resize,
                                                 const float* __restrict__ nv,
                                                 float* __restrict__ cvec) {
    int idx = blockIdx.x * 256 + threadIdx.x;   // 0 .. 2047
    int b = idx >> 7, k = idx & 127;
    const float* wrow = Wr + k * (3 * MD);
    float s = b_resize[k];
    #pragma unroll 8
    for (int m = 0; m < MD; ++m)
        s += wrow[m] * b_e2m[m] + wrow[MD + m] * b_n2m[m] + wrow[2 * MD + m] * nv[b * MD + m];
    cvec[idx] = s;
}

// ---------------- main GEMM kernel ----------------
// out[b][k][n] = sum_f A1[k][f]*e_wv[b][f][n] + sum_f A2[k][f]*h_w[b][f][n] + cvec[b][k]
// Block: 256 threads (8 waves). grid = (N/NT, B).
// LDS holds the X tiles transposed to [n][k] (k-stride SK) so a WMMA B-fragment
// is a single conflict-free ds_load_b64 per lane.
__global__ __launch_bounds__(256) void msg_main(const float* __restrict__ e_wv,
                                                const float* __restrict__ h_w,
                                                const float* __restrict__ A1,
                                                const float* __restrict__ A2,
                                                const float* __restrict__ cvec,
                                                float* __restrict__ out) {
    extern __shared__ float smem[];             // [2][NT][SK] floats = 84480 B

    const int b  = blockIdx.y;
    const int n0 = blockIdx.x * NT;
    const int tid = threadIdx.x;

    // ---- stage e_wv / h_w tiles into LDS, transposed [n][k] ----
    const size_t xbase = (size_t)b * FF * NN + n0;
    #pragma unroll
    for (int i = 0; i < 20; ++i) {
        int c   = i * 256 + tid;                // 0 .. 5119
        int a   = c / 2560;                     // which array
        int rem = c - a * 2560;
        int k   = rem / 20;
        int n4  = rem - k * 20;                 // float4 column chunk
        const float* src = (a ? h_w : e_wv) + xbase + (size_t)k * NN + n4 * 4;
        v4f v = *reinterpret_cast<const v4f*>(src);
        float* dst = smem + a * (NT * SK) + (n4 * 4) * SK + k;
        dst[0 * SK] = v.x; dst[1 * SK] = v.y; dst[2 * SK] = v.z; dst[3 * SK] = v.w;
    }
    __syncthreads();

    const int wave = tid >> 5;
    const int lane = tid & 31;
    const int l15  = lane & 15;
    const int half = lane >> 4;                 // 0: lanes 0-15, 1: lanes 16-31
    const int kb   = half << 1;                 // K sub-offset 0 or 2
    const int wrow = wave * 16;                 // this wave's M strip

    // ---- init accumulators with the broadcast per-batch bias ----
    v8f acc[NT / 16];
    {
        const float* cv = cvec + b * MD + wrow + half * 8;
        v8f ci;
        #pragma unroll
        for (int j = 0; j < 8; ++j) ci[j] = cv[j];
        #pragma unroll
        for (int t = 0; t < NT / 16; ++t) acc[t] = ci;
    }

    // A fragment pointers: lane holds row (wrow + l15), k-pair at (4s + kb)
    const float* a1p = A1 + (wrow + l15) * FF + kb;
    const float* a2p = A2 + (wrow + l15) * FF + kb;
    // B fragment pointers in LDS: lane holds col l15, k-pair at (4s + kb)
    const float* lbe = smem + l15 * SK + kb;
    const float* lbh = smem + NT * SK + l15 * SK + kb;

    #pragma unroll
    for (int s = 0; s < FF / 4; ++s) {
        v2f ae = *reinterpret_cast<const v2f*>(a1p + 4 * s);
        v2f ah = *reinterpret_cast<const v2f*>(a2p + 4 * s);
        #pragma unroll
        for (int t = 0; t < NT / 16; ++t) {
            v2f be = *reinterpret_cast<const v2f*>(lbe + t * 16 * SK + 4 * s);
            acc[t] = __builtin_amdgcn_wmma_f32_16x16x4_f32(
                         false, ae, false, be, (short)0, acc[t], false, false);
            v2f bh = *reinterpret_cast<const v2f*>(lbh + t * 16 * SK + 4 * s);
            acc[t] = __builtin_amdgcn_wmma_f32_16x16x4_f32(
                         false, ah, false, bh, (short)0, acc[t], false, false);
        }
    }

    // ---- store: C/D layout -> VGPR j holds rows wrow+j (lanes 0-15) / wrow+j+8 (lanes 16-31)
    #pragma unroll
    for (int t = 0; t < NT / 16; ++t) {
        const size_t col = (size_t)n0 + t * 16 + l15;
        #pragma unroll
        for (int j = 0; j < 8; ++j) {
            const size_t row = wrow + j + half * 8;
            out[((size_t)b * MD + row) * NN + col] = acc[t][j];
        }
    }
}

// ---------------- launcher ----------------
extern "C" void kernel_launch(void* const* d_in, const int* in_sizes, int n_in,
                              void* d_out, int out_size, void* d_ws, size_t ws_size,
                              hipStream_t stream) {
    const float* h_w      = (const float*)d_in[0];
    const float* h_v      = (const float*)d_in[1];
    const float* e_wv     = (const float*)d_in[2];
    const float* W_e2m    = (const float*)d_in[3];
    const float* b_e2m    = (const float*)d_in[4];
    const float* W_n2m    = (const float*)d_in[5];
    const float* b_n2m    = (const float*)d_in[6];
    const float* W_resize = (const float*)d_in[7];
    const float* b_resize = (const float*)d_in[8];
    float* out = (float*)d_out;

    // workspace layout (floats): A1[16384] | A2[16384] | nv[2048] | cvec[2048]  (~148 KB)
    float* ws   = (float*)d_ws;
    float* A1   = ws;
    float* A2   = ws + 16384;
    float* nv   = ws + 32768;
    float* cvec = ws + 34816;

    prep_A   <<<128, 256, 0, stream>>>(W_resize, W_e2m, W_n2m, A1, A2);
    prep_nv  <<<8,   256, 0, stream>>>(h_v, W_n2m, b_n2m, nv);
    prep_cvec<<<8,   256, 0, stream>>>(W_resize, b_e2m, b_n2m, b_resize, nv, cvec);

    dim3 grid(NN / NT, BB);
    size_t ldsBytes = 2u * NT * SK * sizeof(float);   // 84480 B
    msg_main <<<grid, 256, ldsBytes, stream>>>(e_wv, h_w, A1, A2, cvec, out);
}